// KernelConv_88347477279316
// MI455X (gfx1250) — compile-verified
//
#include <hip/hip_runtime.h>
#include <stdint.h>

typedef float v2f __attribute__((ext_vector_type(2)));
typedef float v4f __attribute__((ext_vector_type(4)));
typedef float v8f __attribute__((ext_vector_type(8)));

// GCC-vector int4 with explicit address spaces, matching the async builtin's
// expected parameter types ('__vector_size__(4*sizeof(int)) int __device__ *').
typedef int i4v __attribute__((vector_size(16)));
typedef __attribute__((address_space(1))) i4v gi4;   // global
typedef __attribute__((address_space(3))) i4v li4;   // LDS

// Problem constants (fixed by the reference setup)
#define BS  4
#define NB  8
#define HH  256
#define WW  256
#define KK  25   // 5x5 taps
#define ROW_V4 1600   // 256 px * 25 taps = 6400 floats = 1600 v4f (16B-aligned)

// ---- CDNA5 async global->LDS copy (ASYNCcnt path), builtin or asm fallback ----
static __device__ __forceinline__ void async_copy_b128(const void* gptr, void* lptr) {
#if defined(__has_builtin) && __has_builtin(__builtin_amdgcn_global_load_async_to_lds_b128)
    __builtin_amdgcn_global_load_async_to_lds_b128((gi4*)gptr, (li4*)lptr, 0, 0);
#else
    unsigned lofs = (unsigned)(uintptr_t)(__attribute__((address_space(3))) void*)lptr;
    asm volatile("global_load_async_to_lds_b128 %0, %1, off"
                 :: "v"(lofs), "v"(gptr) : "memory");
#endif
}

static __device__ __forceinline__ void wait_asynccnt0() {
#if defined(__has_builtin) && __has_builtin(__builtin_amdgcn_s_wait_asynccnt)
    __builtin_amdgcn_s_wait_asynccnt(0);
#else
    asm volatile("s_wait_asynccnt 0x0" ::: "memory");
#endif
}

// ---------------------------------------------------------------------------
// Kernel 1: per-pixel dynamic 5x5 filtering -> pred_i  (the 200 MB stream)
// One 256-thread block per (b, n, y) row. Core row slab is DMA'd straight
// into LDS with async b128 copies while the wave stages the 5 frame rows.
// ---------------------------------------------------------------------------
__global__ __launch_bounds__(256) void kpn_filter_kernel(
    const float* __restrict__ frames,   // (bs, N, h, w)
    const float* __restrict__ core,     // flat; row slab is contiguous
    float* __restrict__ pred_i)         // (bs, N, h, w)
{
    __shared__ v4f   score4[ROW_V4];    // 25.6 KB: weights for this row
    __shared__ float sfr[5][WW + 4];    // 5 frame rows with 2-wide zero halo

    const int tid  = threadIdx.x;
    const int sblk = blockIdx.x;        // 0 .. 8191
    const int y = sblk & 255;
    const int n = (sblk >> 8) & 7;
    const int b = sblk >> 11;

    // ---- async-stage core row: float base = b*(h*w*200) + n*(h*w*25) + y*(w*25)
    const v4f* gc4 = (const v4f*)core + (b * 3276800 + n * 409600 + y * 1600);
    #pragma unroll
    for (int i = 0; i < 7; ++i) {
        int idx = tid + i * 256;
        if (idx < ROW_V4) async_copy_b128(gc4 + idx, &score4[idx]);
    }

    // ---- stage 5 frame rows with zero padding (normal path: needs zero-fill,
    //      and frames are 25x-reused so we want them resident in L2) ----
    if (tid < 20) {                     // zero the 4 halo cells of each row
        int r = tid >> 2, c = tid & 3;
        sfr[r][(c < 2) ? c : (c + 256)] = 0.0f;
    }
    const float* fbase = frames + ((b * NB + n) << 16);
    #pragma unroll
    for (int r = 0; r < 5; ++r) {
        int row = y + r - 2;
        sfr[r][tid + 2] = (row >= 0 && row < HH) ? fbase[(row << 8) + tid] : 0.0f;
    }

    wait_asynccnt0();                   // my async LDS writes have landed
    __syncthreads();                    // ...and are published to the workgroup

    // ---- 25-tap dot product (LDS: stride-25 dwords across lanes = conflict-free) ----
    const float* wv = (const float*)score4 + tid * KK;
    float acc = 0.0f;
    #pragma unroll
    for (int ky = 0; ky < 5; ++ky)
        #pragma unroll
        for (int kx = 0; kx < 5; ++kx)
            acc = fmaf(wv[ky * 5 + kx], sfr[ky][tid + kx], acc);

    pred_i[((b * NB + n) << 16) + (y << 8) + tid] = acc;
}

// ---------------------------------------------------------------------------
// Kernel 2: mean over the burst dimension via V_WMMA_F32_16X16X4_F32.
// D = A x B + C with A = [16 pixels] x [K=4 burst frames] (two K-steps => N=8),
// B = 0.125 everywhere (layout-independent!), so every column of D is the mean.
// ---------------------------------------------------------------------------
__global__ __launch_bounds__(256) void burst_mean_wmma(
    const float* __restrict__ pred_i,   // (bs, N, h, w)
    float* __restrict__ pred_img)       // (bs, h, w)
{
    const int tid  = threadIdx.x;
    const int lane = tid & 31;
    const int wv   = tid >> 5;
    const int pixbase = blockIdx.x * 128 + wv * 16;   // 16 pixels per wave
    const int p   = pixbase + (lane & 15);            // this lane's A-row pixel
    const int b   = p >> 16;
    const int rem = p & 65535;                        // y*256 + x

    const float* src = pred_i + (b << 19) + rem;      // (b*8)<<16, n-stride 65536
    const int koff = (lane < 16) ? 0 : 2;             // ISA A-layout: lanes16-31 hold K=2,3

    v2f a0, a1;
    a0.x = src[(koff + 0) << 16];
    a0.y = src[(koff + 1) << 16];
    a1.x = src[(koff + 4) << 16];
    a1.y = src[(koff + 5) << 16];

    const v2f bv = {0.125f, 0.125f};
    v8f c = {};

    // (neg_a, A, neg_b, B, c_mod, C, reuse_a, reuse_b)
    c = __builtin_amdgcn_wmma_f32_16x16x4_f32(false, a0, false, bv, (short)0, c, false, false);
    c = __builtin_amdgcn_wmma_f32_16x16x4_f32(false, a1, false, bv, (short)0, c, false, false);

    // C/D layout: VGPR r -> lanes0-15: M=r (N=lane), lanes16-31: M=r+8.
    // All columns equal, so lane 0 holds means for pixels +0..7, lane 16 for +8..15.
    if (lane == 0 || lane == 16) {
        float* o = pred_img + pixbase + ((lane >> 4) << 3);
        #pragma unroll
        for (int r = 0; r < 8; ++r) o[r] = c[r];
    }
}

// ---------------------------------------------------------------------------
extern "C" void kernel_launch(void* const* d_in, const int* in_sizes, int n_in,
                              void* d_out, int out_size, void* d_ws, size_t ws_size,
                              hipStream_t stream)
{
    const float* frames = (const float*)d_in[0];   // 4*8*256*256 f32
    const float* core   = (const float*)d_in[1];   // 4*256*256*200 f32

    float* out      = (float*)d_out;
    float* pred_img = out;                          // 4*256*256       (first output)
    float* pred_i   = out + BS * HH * WW;           // 4*8*256*256     (second output)

    kpn_filter_kernel<<<BS * NB * HH, 256, 0, stream>>>(frames, core, pred_i);
    burst_mean_wmma<<<(BS * HH * WW) / 128, 256, 0, stream>>>(pred_i, pred_img);
}